// SelfAttention_60730837566374
// MI455X (gfx1250) — compile-verified
//
#include <hip/hip_runtime.h>
#include <math.h>

// ---------------------------------------------------------------------------
// Problem constants (match reference)
// ---------------------------------------------------------------------------
static constexpr int Bc  = 2;
static constexpr int Tc  = 2048;
static constexpr int Dc  = 1024;
static constexpr int Hc  = 16;
static constexpr int HDc = 64;     // head dim == ROPE -> whole head rotated

typedef float v2f __attribute__((ext_vector_type(2)));
typedef float v8f __attribute__((ext_vector_type(8)));
typedef int   v4i __attribute__((ext_vector_type(4)));

// f32 WMMA: D(16x16,f32) = A(16x4,f32) * B(4x16,f32) + C
#define WMMA_F32(a, b, c) \
    __builtin_amdgcn_wmma_f32_16x16x4_f32(false, (a), false, (b), (short)0, (c), false, false)

// ---------------------------------------------------------------------------
// Async global->LDS staging (CDNA5 GLOBAL_LOAD_ASYNC_TO_LDS_B128, ASYNCcnt)
// ---------------------------------------------------------------------------
#ifndef __has_builtin
#define __has_builtin(x) 0
#endif
#if __has_builtin(__builtin_amdgcn_global_load_async_to_lds_b128)
#define HAVE_ASYNC_LDS 1
#else
#define HAVE_ASYNC_LDS 0
#endif

#define AS1 __attribute__((address_space(1)))
#define AS3 __attribute__((address_space(3)))

__device__ __forceinline__ void async_copy_b128(const float* g, float* l) {
#if HAVE_ASYNC_LDS
    __builtin_amdgcn_global_load_async_to_lds_b128(
        (AS1 v4i*)g, (AS3 v4i*)l, 0, 0);
#else
    *(float4*)l = *(const float4*)g;   // sync fallback
#endif
}

#if HAVE_ASYNC_LDS
#define ASYNC_WAIT(n) asm volatile("s_wait_asynccnt " #n ::: "memory")
#else
#define ASYNC_WAIT(n) do { } while (0)
#endif

// ---------------------------------------------------------------------------
// Stage 1: qkv = x @ Wqkv^T + b ; RoPE on q,k ; q pre-scaled by 1/sqrt(HD)
// Output layout: Q/K/V as [B, H, T, HD]
// Workgroup: 128 threads (4 waves). Tile: 64(M) x 64(N). Each wave: 16 x 64.
// Double-buffered async LDS staging overlaps the fill with the WMMA pipe.
// ---------------------------------------------------------------------------
__global__ __launch_bounds__(128) void qkv_rope_kernel(
    const float* __restrict__ x, const float* __restrict__ W,
    const float* __restrict__ bias,
    float* __restrict__ Qb, float* __restrict__ Kb, float* __restrict__ Vb)
{
    constexpr int BM = 64, BN = 64, BK = 16, PAD = 20; // 80B rows, 16B aligned
    __shared__ float lA[2][BM * PAD];
    __shared__ float lB[2][BN * PAD];

    const int tn   = (3 * Dc) / BN;                 // 48 column tiles
    const int m0   = (int)(blockIdx.x / tn) * BM;
    const int n0   = (int)(blockIdx.x % tn) * BN;
    const int tid  = threadIdx.x;
    const int lane = tid & 31;
    const int wave = tid >> 5;
    const int lg   = lane >> 4;                     // lane group (K-half)
    const int lm   = lane & 15;                     // M/N index within fragment
    const int koff = lg * 2;

    auto stage = [&](int k0, int buf) {             // 4 async b128 per thread
        #pragma unroll
        for (int q = 0; q < 2; ++q) {
            int slot = tid + q * 128;               // 0..255
            int row  = slot >> 2;
            int kq   = (slot & 3) << 2;
            async_copy_b128(x + (size_t)(m0 + row) * Dc + k0 + kq,
                            &lA[buf][row * PAD + kq]);
            async_copy_b128(W + (size_t)(n0 + row) * Dc + k0 + kq,
                            &lB[buf][row * PAD + kq]);
        }
    };

    v8f acc[4] = {};
    stage(0, 0);

    for (int k0 = 0; k0 < Dc; k0 += BK) {
        const int  buf  = (k0 / BK) & 1;
        const bool more = (k0 + BK) < Dc;
        if (more) {
            stage(k0 + BK, buf ^ 1);
            ASYNC_WAIT(4);                          // chunk k complete (in-order)
        } else {
            ASYNC_WAIT(0);
        }
        __syncthreads();

        const float* ap = lA[buf] + (wave * 16 + lm) * PAD + koff;
        const float* bp = lB[buf] + lm * PAD + koff;
        #pragma unroll
        for (int j = 0; j < 4; ++j) {               // 4 k-steps of 4
            v2f af = *(const v2f*)(ap + 4 * j);
            #pragma unroll
            for (int s = 0; s < 4; ++s) {           // 4 n-subtiles
                v2f bf = *(const v2f*)(bp + s * 16 * PAD + 4 * j);
                acc[s] = WMMA_F32(af, bf, acc[s]);
            }
        }
        __syncthreads();                            // reads done before reuse
    }

    // ---- epilogue: bias, RoPE, scatter ----
    const int which = n0 / Dc;              // 0=q 1=k 2=v
    const int h     = (n0 % Dc) / HDc;      // head
    const int mbase = m0 + wave * 16 + 8 * lg;

    #pragma unroll
    for (int s = 0; s < 4; ++s) {
        float bv = bias[n0 + s * 16 + lm];
        #pragma unroll
        for (int e = 0; e < 8; ++e) acc[s][e] += bv;
    }

    if (which < 2) {
        // pair (d, d+32): acc[s] <-> acc[s+2], d = s*16 + lm in 0..31
        #pragma unroll
        for (int s = 0; s < 2; ++s) {
            // inv_freq = 10000^(-d/32) = exp2(-d * log2(10000)/32)
            float inv = exp2f(-(float)(s * 16 + lm) * (13.287712379549449f / 32.0f));
            #pragma unroll
            for (int e = 0; e < 8; ++e) {
                int   row = mbase + e;
                float t   = (float)(row & (Tc - 1));
                float th  = t * inv;
                float cth = cosf(th), sth = sinf(th);
                float x1 = acc[s][e], x2 = acc[s + 2][e];
                acc[s][e]     = x1 * cth - x2 * sth;
                acc[s + 2][e] = x1 * sth + x2 * cth;
            }
        }
    }
    const float qscale = (which == 0) ? 0.125f : 1.0f;  // fold 1/sqrt(64) into Q
    float* dst = (which == 0) ? Qb : ((which == 1) ? Kb : Vb);
    #pragma unroll
    for (int e = 0; e < 8; ++e) {
        int row = mbase + e;
        int bb  = row >> 11;                // row / T
        int t   = row & (Tc - 1);
        size_t base = (((size_t)(bb * Hc + h)) * Tc + t) * HDc;
        #pragma unroll
        for (int s = 0; s < 4; ++s)
            dst[base + s * 16 + lm] = acc[s][e] * qscale;
    }
}

// ---------------------------------------------------------------------------
// Stage 2: flash attention. One wave per 16-query tile; 4 waves per block.
// allow(q,k) = (k <= q) || is_global(q);  is_global <=> 2 <= input_id <= 7.
// K tiles: per-wave async double-buffered LDS staging (ASYNCcnt is per-wave,
// so no barriers are needed at all).
// ---------------------------------------------------------------------------
__global__ __launch_bounds__(128) void attn_kernel(
    const float* __restrict__ Qb, const float* __restrict__ Kb,
    const float* __restrict__ Vb, const int* __restrict__ ids,
    float* __restrict__ Ob)
{
    constexpr int KPAD = 68;   // 272B rows: 16B aligned
    constexpr int PPAD = 20;
    __shared__ float lK[4][2][16 * KPAD];
    __shared__ float lP[4][16 * PPAD];

    const int tid = threadIdx.x, lane = tid & 31, wave = tid >> 5;
    const int lg = lane >> 4, lm = lane & 15, koff = lg * 2;

    const int qt  = (int)blockIdx.x * 4 + wave;  // global q-tile id
    const int qtl = qt & 127;                    // 128 q-tiles per (b,h)
    const int bh  = qt >> 7;
    const int b   = bh >> 4;
    const int h   = bh & 15;
    const int q0  = qtl * 16;

    const float* Qp = Qb + (size_t)bh * Tc * HDc;
    const float* Kp = Kb + (size_t)bh * Tc * HDc;
    const float* Vp = Vb + (size_t)bh * Tc * HDc;
    float* lPw = lP[wave];

    auto stageK = [&](int kt, int buf) {         // 8 async b128 per lane
        const int k0 = kt * 16;
        float* dst = lK[wave][buf];
        #pragma unroll
        for (int q = 0; q < 8; ++q) {
            int slot = lane + q * 32;            // 0..255 float4 slots
            int row  = slot >> 4;
            int c4   = (slot & 15) << 2;
            async_copy_b128(Kp + (size_t)(k0 + row) * HDc + c4,
                            &dst[row * KPAD + c4]);
        }
    };

    // Preload Q A-fragments for the whole q-tile (16 k-steps over HD=64)
    v2f qa[16];
    #pragma unroll
    for (int j = 0; j < 16; ++j)
        qa[j] = *(const v2f*)(Qp + (size_t)(q0 + lm) * HDc + 4 * j + koff);

    // Per-row "global token" flags (query-side mask)
    bool gq[8];
    bool anyl = false;
    #pragma unroll
    for (int e = 0; e < 8; ++e) {
        int id = ids[b * Tc + q0 + e + 8 * lg];
        gq[e]  = (id >= 2) && (id <= 7);
        anyl  |= gq[e];
    }
    const bool anyGlobal = (__any((int)anyl) != 0);
    const int  nkt = anyGlobal ? (Tc / 16) : (qtl + 1);

    float m[8], l[8];
    v8f accO[4] = {};
    #pragma unroll
    for (int e = 0; e < 8; ++e) { m[e] = -3.0e38f; l[e] = 0.0f; }

    stageK(0, 0);

    for (int kt = 0; kt < nkt; ++kt) {
        const int  k0   = kt * 16;
        const int  buf  = kt & 1;
        const bool more = (kt + 1) < nkt;        // wave-uniform
        if (more) {
            stageK(kt + 1, buf ^ 1);
            ASYNC_WAIT(8);                       // tile kt landed (in-order)
        } else {
            ASYNC_WAIT(0);
        }
        const float* lKw = lK[wave][buf];

        // S = Q @ K^T : B-frag lane->kpos, contraction over d
        v8f sacc = {};
        #pragma unroll
        for (int j = 0; j < 16; ++j) {
            v2f bf = *(const v2f*)(lKw + lm * KPAD + 4 * j + koff);
            sacc = WMMA_F32(qa[j], bf, sacc);
        }

        // mask + online softmax (rows live across 16-lane halves)
        const int kpos = k0 + lm;
        float p[8], alpha[8];
        #pragma unroll
        for (int e = 0; e < 8; ++e) {
            int   qrow = q0 + e + 8 * lg;
            float sv = sacc[e];
            if (!((kpos <= qrow) || gq[e])) sv = -3.0e38f;
            float r = sv;
            r = fmaxf(r, __shfl_xor(r, 1));
            r = fmaxf(r, __shfl_xor(r, 2));
            r = fmaxf(r, __shfl_xor(r, 4));
            r = fmaxf(r, __shfl_xor(r, 8));
            float mn = fmaxf(m[e], r);
            alpha[e] = __expf(m[e] - mn);
            float pv = __expf(sv - mn);
            float ps = pv;
            ps += __shfl_xor(ps, 1);
            ps += __shfl_xor(ps, 2);
            ps += __shfl_xor(ps, 4);
            ps += __shfl_xor(ps, 8);
            l[e] = l[e] * alpha[e] + ps;
            m[e] = mn;
            p[e] = pv;
        }
        #pragma unroll
        for (int s = 0; s < 4; ++s)
            #pragma unroll
            for (int e = 0; e < 8; ++e) accO[s][e] *= alpha[e];

        // C-layout P -> LDS -> A-fragment layout (wave-local round trip)
        #pragma unroll
        for (int e = 0; e < 8; ++e)
            lPw[(e + 8 * lg) * PPAD + lm] = p[e];
        asm volatile("s_wait_dscnt 0" ::: "memory");

        // O += P @ V  (V B-frags direct from global: 16 consecutive d per
        // half-wave -> 64B coalesced)
        #pragma unroll
        for (int j = 0; j < 4; ++j) {
            v2f af = *(const v2f*)(lPw + lm * PPAD + 4 * j + koff);
            int kr = k0 + 4 * j + koff;
            #pragma unroll
            for (int s = 0; s < 4; ++s) {
                v2f bf;
                bf.x = Vp[(size_t)kr * HDc + s * 16 + lm];
                bf.y = Vp[(size_t)(kr + 1) * HDc + s * 16 + lm];
                accO[s] = WMMA_F32(af, bf, accO[s]);
            }
        }
    }

    // normalize + scatter to [B, T, H*HD]
    #pragma unroll
    for (int e = 0; e < 8; ++e) {
        int   qrow = q0 + e + 8 * lg;
        float invl = 1.0f / l[e];
        size_t base = ((size_t)(b * Tc + qrow)) * (Hc * HDc) + h * HDc;
        #pragma unroll
        for (int s = 0; s < 4; ++s)
            Ob[base + s * 16 + lm] = accO[s][e] * invl;
    }
}

// ---------------------------------------------------------------------------
// Stage 3: out = O @ Wout^T + b_out.  Same double-buffered tiling, no RoPE.
// ---------------------------------------------------------------------------
__global__ __launch_bounds__(128) void outproj_kernel(
    const float* __restrict__ O, const float* __restrict__ W,
    const float* __restrict__ bias, float* __restrict__ out)
{
    constexpr int BM = 64, BN = 64, BK = 16, PAD = 20;
    __shared__ float lA[2][BM * PAD];
    __shared__ float lB[2][BN * PAD];

    const int tn   = Dc / BN;                       // 16 column tiles
    const int m0   = (int)(blockIdx.x / tn) * BM;
    const int n0   = (int)(blockIdx.x % tn) * BN;
    const int tid  = threadIdx.x;
    const int lane = tid & 31;
    const int wave = tid >> 5;
    const int lg   = lane >> 4;
    const int lm   = lane & 15;
    const int koff = lg * 2;

    auto stage = [&](int k0, int buf) {
        #pragma unroll
        for (int q = 0; q < 2; ++q) {
            int slot = tid + q * 128;
            int row  = slot >> 2;
            int kq   = (slot & 3) << 2;
            async_copy_b128(O + (size_t)(m0 + row) * Dc + k0 + kq,
                            &lA[buf][row * PAD + kq]);
            async_copy_b128(W + (size_t)(n0 + row) * Dc + k0 + kq,
                            &lB[buf][row * PAD + kq]);
        }
    };

    v8f acc[4] = {};
    stage(0, 0);

    for (int k0 = 0; k0 < Dc; k0 += BK) {
        const int  buf  = (k0 / BK) & 1;
        const bool more = (k0 + BK) < Dc;
        if (more) {
            stage(k0 + BK, buf ^ 1);
            ASYNC_WAIT(4);
        } else {
            ASYNC_WAIT(0);
        }
        __syncthreads();

        const float* ap = lA[buf] + (wave * 16 + lm) * PAD + koff;
        const float* bp = lB[buf] + lm * PAD + koff;
        #pragma unroll
        for (int j = 0; j < 4; ++j) {
            v2f af = *(const v2f*)(ap + 4 * j);
            #pragma unroll
            for (int s = 0; s < 4; ++s) {
                v2f bf = *(const v2f*)(bp + s * 16 * PAD + 4 * j);
                acc[s] = WMMA_F32(af, bf, acc[s]);
            }
        }
        __syncthreads();
    }

    const int mbase = m0 + wave * 16 + 8 * lg;
    #pragma unroll
    for (int s = 0; s < 4; ++s) {
        float bv = bias[n0 + s * 16 + lm];
        #pragma unroll
        for (int e = 0; e < 8; ++e) {
            int row = mbase + e;
            out[(size_t)row * Dc + n0 + s * 16 + lm] = acc[s][e] + bv;
        }
    }
}

// ---------------------------------------------------------------------------
// Launch: ws layout = Q | K | V (each [B,H,T,HD]) | O ([B,T,H*HD]) = 4x16MB
// ---------------------------------------------------------------------------
extern "C" void kernel_launch(void* const* d_in, const int* in_sizes, int n_in,
                              void* d_out, int out_size, void* d_ws, size_t ws_size,
                              hipStream_t stream) {
    (void)in_sizes; (void)n_in; (void)out_size; (void)ws_size;
    const float* x    = (const float*)d_in[0];
    const int*   ids  = (const int*)  d_in[1];
    const float* Wqkv = (const float*)d_in[2];
    const float* bqkv = (const float*)d_in[3];
    const float* Wout = (const float*)d_in[4];
    const float* bout = (const float*)d_in[5];
    float* out = (float*)d_out;

    const size_t nBHTD = (size_t)Bc * Hc * Tc * HDc;   // 4M floats
    float* Qb = (float*)d_ws;
    float* Kb = Qb + nBHTD;
    float* Vb = Kb + nBHTD;
    float* Ob = Vb + nBHTD;

    // Stage 1: (B*T/64) * (3D/64) = 64 * 48 blocks
    qkv_rope_kernel<<<dim3((Bc * Tc / 64) * ((3 * Dc) / 64)), dim3(128), 0, stream>>>(
        x, Wqkv, bqkv, Qb, Kb, Vb);
    // Stage 2: (B*H*T/16)/4 = 1024 blocks
    attn_kernel<<<dim3((Bc * Hc * (Tc / 16)) / 4), dim3(128), 0, stream>>>(
        Qb, Kb, Vb, ids, Ob);
    // Stage 3: (B*T/64) * (D/64) = 64 * 16 blocks
    outproj_kernel<<<dim3((Bc * Tc / 64) * (Dc / 64)), dim3(128), 0, stream>>>(
        Ob, Wout, bout, out);
}